// AdaptedCELoss_41566693491288
// MI455X (gfx1250) — compile-verified
//
#include <hip/hip_runtime.h>
#include <hip/hip_bf16.h>
#include <math.h>

typedef __attribute__((ext_vector_type(2))) float v2f;
typedef __attribute__((ext_vector_type(8))) float v8f;

#define ROW_LEN 4096
#define THREADS 256

// One workgroup per row. Each thread owns 16 row elements in registers
// (4 coalesced b128 loads of logits + 4 of targets), then 3 reduction phases.
__global__ __launch_bounds__(THREADS) void
rowwise_kernel(const float* __restrict__ logits,
               const int* __restrict__ targets,
               float* __restrict__ rtot,
               float* __restrict__ rcnt)
{
    const int t = threadIdx.x;
    const int wid  = t >> 5;
    const int lane = t & 31;
    const long long row = blockIdx.x;

    const float4* lp = (const float4*)(logits + row * (long long)ROW_LEN);
    const int4*   tp = (const int4*)(targets + row * (long long)ROW_LEN);

    float x[16];
    unsigned posm = 0u, negm = 0u;
#pragma unroll
    for (int j = 0; j < 4; ++j) {
        float4 a = lp[j * THREADS + t];   // lane stride 16B -> coalesced b128
        int4   b = tp[j * THREADS + t];
        x[4*j+0] = a.x; x[4*j+1] = a.y; x[4*j+2] = a.z; x[4*j+3] = a.w;
        posm |= ((b.x == 1) ? 1u : 0u) << (4*j+0);
        posm |= ((b.y == 1) ? 1u : 0u) << (4*j+1);
        posm |= ((b.z == 1) ? 1u : 0u) << (4*j+2);
        posm |= ((b.w == 1) ? 1u : 0u) << (4*j+3);
        negm |= ((b.x == 0) ? 1u : 0u) << (4*j+0);
        negm |= ((b.y == 0) ? 1u : 0u) << (4*j+1);
        negm |= ((b.z == 0) ? 1u : 0u) << (4*j+2);
        negm |= ((b.w == 0) ? 1u : 0u) << (4*j+3);
    }

    __shared__ float red[8];
    __shared__ float redc[8];

    // ---- Phase A: row max over negatives (branchless select) ----
    float m = -INFINITY;
#pragma unroll
    for (int k = 0; k < 16; ++k)
        m = fmaxf(m, ((negm >> k) & 1u) ? x[k] : -INFINITY);
#pragma unroll
    for (int off = 16; off > 0; off >>= 1)
        m = fmaxf(m, __shfl_xor(m, off, 32));
    if (lane == 0) red[wid] = m;
    __syncthreads();
    float rmax = red[0];
#pragma unroll
    for (int w = 1; w < 8; ++w) rmax = fmaxf(rmax, red[w]);
    __syncthreads();

    // ---- Phase B: sum exp(x - rmax) over negatives ----
    float s = 0.0f;
#pragma unroll
    for (int k = 0; k < 16; ++k) {
        float e = expf(x[k] - rmax);
        s += ((negm >> k) & 1u) ? e : 0.0f;
    }
#pragma unroll
    for (int off = 16; off > 0; off >>= 1)
        s += __shfl_xor(s, off, 32);
    if (lane == 0) red[wid] = s;
    __syncthreads();
    float sumexp = red[0] + red[1] + red[2] + red[3]
                 + red[4] + red[5] + red[6] + red[7];
    // no negatives in row: rmax = -inf, sumexp = 0 -> neg_lse = -inf
    float neg_lse = rmax + logf(sumexp);
    __syncthreads();

    // ---- Phase C: sum softplus(neg_lse - x) over positives + count ----
    float tot = 0.0f;
#pragma unroll
    for (int k = 0; k < 16; ++k) {
        float d  = neg_lse - x[k];
        float sp = fmaxf(d, 0.0f) + log1pf(expf(-fabsf(d))); // softplus(-inf)=0
        tot += ((posm >> k) & 1u) ? sp : 0.0f;
    }
    float c = (float)__popc(posm); // <= 16, exact
#pragma unroll
    for (int off = 16; off > 0; off >>= 1) {
        tot += __shfl_xor(tot, off, 32);
        c   += __shfl_xor(c,   off, 32);
    }
    if (lane == 0) { red[wid] = tot; redc[wid] = c; }
    __syncthreads();
    if (t == 0) {
        float T = red[0]+red[1]+red[2]+red[3]+red[4]+red[5]+red[6]+red[7];
        float C = redc[0]+redc[1]+redc[2]+redc[3]+redc[4]+redc[5]+redc[6]+redc[7];
        rtot[row] = T;
        rcnt[row] = C;   // exact integer <= 4096 in fp32
    }
}

// Single-workgroup finalize. Wave 0 sums the per-row totals with
// V_WMMA_F32_16X16X4_F32 (A = ones, B = data, C accumulates); counts are
// reduced in int32 (num_pos ~ 33.5M exceeds fp32's 2^24 exact range).
__global__ __launch_bounds__(THREADS) void
finalize_kernel(const float* __restrict__ rtot,
                const float* __restrict__ rcnt,
                float* __restrict__ out,
                int nrows)
{
    const int t = threadIdx.x;
    const int wid  = t >> 5;
    const int lane = t & 31;

    // integer count reduction across all 256 threads
    int c = 0;
    for (int i = t; i < nrows; i += THREADS) c += (int)rcnt[i];
#pragma unroll
    for (int off = 16; off > 0; off >>= 1) c += __shfl_xor(c, off, 32);
    __shared__ int ic[8];
    if (lane == 0) ic[wid] = c;

    // WMMA total reduction on wave 0 (EXEC all-ones within the wave)
    float total = 0.0f;
    if (t < 32) {
        v8f acc = {};                 // C/D accumulator, 16x16 f32
        v2f a;  a.x = 1.0f; a.y = 1.0f;   // A = 16x4 ones
        const int main_n = nrows & ~63;
        for (int i = 0; i < main_n; i += 64) {
            v2f b;                    // B = 4x16 data tile (64 values/iter)
            b.x = rtot[i + t];
            b.y = rtot[i + 32 + t];
            acc = __builtin_amdgcn_wmma_f32_16x16x4_f32(
                      false, a, false, b, (short)0, acc, false, false);
        }
        // D row M=0 lives in acc[0], lanes 0..15: sum those 16 lanes
        float r0 = acc[0];
#pragma unroll
        for (int off = 8; off > 0; off >>= 1) r0 += __shfl_xor(r0, off, 32);
        // scalar tail (nrows=16384 -> empty, kept for safety)
        if (t == 0) {
            for (int i = main_n; i < nrows; ++i) r0 += rtot[i];
            total = r0;
        }
    }
    __syncthreads();
    if (t == 0) {
        int cnt = ic[0]+ic[1]+ic[2]+ic[3]+ic[4]+ic[5]+ic[6]+ic[7];
        out[0] = (cnt > 0) ? (total / (float)cnt) : 0.0f;
    }
}

extern "C" void kernel_launch(void* const* d_in, const int* in_sizes, int n_in,
                              void* d_out, int out_size, void* d_ws, size_t ws_size,
                              hipStream_t stream) {
    (void)n_in; (void)out_size; (void)ws_size;
    const float* logits  = (const float*)d_in[0];
    const int*   targets = (const int*)d_in[1];
    const int B = in_sizes[0] / ROW_LEN;   // 16384

    float* rtot = (float*)d_ws;            // [B]
    float* rcnt = rtot + B;                // [B]  (2*B*4 = 128 KB scratch)

    rowwise_kernel<<<B, THREADS, 0, stream>>>(logits, targets, rtot, rcnt);
    finalize_kernel<<<1, THREADS, 0, stream>>>(rtot, rcnt, (float*)d_out, B);
}